// GAT_28432683499971
// MI455X (gfx1250) — compile-verified
//
#include <hip/hip_runtime.h>
#include <math.h>

// ---------------------------------------------------------------------------
// CDNA5 (gfx1250) wave32 WMMA fp32 helpers: V_WMMA_F32_16X16X4_F32
//   A: 16x4  f32, 2 VGPR/lane: lane<16 -> M=lane,   K=k0+{0,1}
//                              lane>=16 -> M=lane-16, K=k0+{2,3}
//   B: 4x16  f32, 2 VGPR/lane: lane<16 -> N=lane,   K=k0+{0,1}
//                              lane>=16 -> N=lane-16, K=k0+{2,3}
//   C/D: 16x16 f32, 8 VGPR: v[i] = (M = i + 8*(lane>>4), N = lane&15)
// ---------------------------------------------------------------------------
typedef float v2f __attribute__((ext_vector_type(2)));
typedef float v8f __attribute__((ext_vector_type(8)));

__device__ __forceinline__ v8f wmma_f32(v2f a, v2f b, v8f c) {
  return __builtin_amdgcn_wmma_f32_16x16x4_f32(false, a, false, b, (short)0, c,
                                               false, false);
}

// One wave computes rows [m0, m0+16) of  out = act(A @ W^T + bias)
// A: [Nrows, K] row-major. W: [OUT, K] row-major. out: ldo-strided.
__device__ void wave_linear_tile(const float* __restrict__ A,
                                 const float* __restrict__ W,
                                 const float* __restrict__ bias,
                                 float* __restrict__ out,
                                 int m0, int Nrows, int K, int OUT, int ldo,
                                 float slope, bool act) {
  const int lane = threadIdx.x & 31;
  const int half = lane >> 4;    // K sub-offset selector
  const int l15  = lane & 15;    // row (A) / col (B,D) within tile
  int mload = m0 + l15;
  if (mload >= Nrows) mload = Nrows - 1;            // clamp (stores guarded)
  const float* arow = A + (size_t)mload * K + half * 2;
  const bool full = (m0 + 16) <= Nrows;

  for (int nb = 0; nb < OUT; nb += 16) {
    const int ncol = nb + l15;
    const float* brow = W + (size_t)ncol * K + half * 2;
    v8f acc = {};
    int k0 = 0;
    // main loop with streaming prefetch 256B ahead (first n-tile pass only;
    // later passes re-hit L0/L2 for A, and W stays cache-resident)
#pragma unroll 8
    for (; k0 + 64 <= K; k0 += 4) {
      if (nb == 0 && (k0 & 31) == 0) __builtin_prefetch(arow + k0 + 64, 0, 3);
      v2f a = *(const v2f*)(arow + k0);
      v2f b = *(const v2f*)(brow + k0);
      acc = wmma_f32(a, b, acc);
    }
#pragma unroll 8
    for (; k0 < K; k0 += 4) {
      v2f a = *(const v2f*)(arow + k0);
      v2f b = *(const v2f*)(brow + k0);
      acc = wmma_f32(a, b, acc);
    }

    const float bb = bias ? bias[ncol] : 0.0f;
    if (full) {
      // fast path: unguarded, 32-bit address stepping; lanes of one half
      // write 16 consecutive floats -> 64B segments
      float* orow = out + (m0 + 8 * half) * ldo + ncol;
#pragma unroll
      for (int i = 0; i < 8; ++i) {
        float v = acc[i] + bb;
        if (act) v = v > 0.0f ? v : slope * v;
        orow[i * ldo] = v;
      }
    } else {
#pragma unroll
      for (int i = 0; i < 8; ++i) {
        const int m = m0 + 8 * half + i;
        if (m < Nrows) {
          float v = acc[i] + bb;
          if (act) v = v > 0.0f ? v : slope * v;
          out[(size_t)m * ldo + ncol] = v;
        }
      }
    }
  }
}

// Small-K projection (K=6 or 11): lane j owns output column j (32 cols).
template <int K>
__device__ void wave_small_proj(const float* __restrict__ X,
                                const float* __restrict__ W,
                                const float* __restrict__ bias,
                                float* __restrict__ out,  // col-offset applied
                                int m0, int Nrows) {
  const int lane = threadIdx.x & 31;
  float w[K];
#pragma unroll
  for (int k = 0; k < K; ++k) w[k] = W[lane * K + k];
  const float b = bias[lane];
#pragma unroll 4
  for (int i = 0; i < 16; ++i) {
    const int m = m0 + i;
    if (m >= Nrows) break;
    float s = b;
#pragma unroll
    for (int k = 0; k < K; ++k) s += X[(size_t)m * K + k] * w[k];
    s = s > 0.0f ? s : 0.01f * s;
    out[(size_t)m * 160 + lane] = s;
  }
}

// ---------------------------------------------------------------------------
// Kernel 1: fused 5-way input projection -> h_cat[N,160] (leaky_relu 0.01)
// ---------------------------------------------------------------------------
__global__ void k_fuse(const float* __restrict__ pre_x,
                       const float* __restrict__ num_prop,
                       const float* __restrict__ num_cat,
                       const float* __restrict__ des,
                       const float* __restrict__ tweet,
                       const float* __restrict__ w_np, const float* __restrict__ b_np,
                       const float* __restrict__ w_nc, const float* __restrict__ b_nc,
                       const float* __restrict__ w_des, const float* __restrict__ b_des,
                       const float* __restrict__ w_text, const float* __restrict__ b_text,
                       const float* __restrict__ w_tw, const float* __restrict__ b_tw,
                       float* __restrict__ h_cat, int Nrows) {
  const int wave = blockIdx.x * (blockDim.x >> 5) + (threadIdx.x >> 5);
  const int m0 = wave * 16;
  if (m0 >= Nrows) return;
  wave_small_proj<6>(num_prop, w_np, b_np, h_cat + 0, m0, Nrows);
  wave_small_proj<11>(num_cat, w_nc, b_nc, h_cat + 32, m0, Nrows);
  wave_linear_tile(des,   w_des,  b_des,  h_cat + 64,  m0, Nrows, 768, 32, 160, 0.01f, true);
  wave_linear_tile(tweet, w_text, b_text, h_cat + 96,  m0, Nrows, 768, 32, 160, 0.01f, true);
  wave_linear_tile(pre_x, w_tw,   b_tw,   h_cat + 128, m0, Nrows, 768, 32, 160, 0.01f, true);
}

// ---------------------------------------------------------------------------
// Kernel 2: generic WMMA linear: out = [lrelu](A @ W^T [+ bias])
// ---------------------------------------------------------------------------
template <bool ACT>
__global__ void k_linear(const float* __restrict__ A, const float* __restrict__ W,
                         const float* __restrict__ bias, float* __restrict__ out,
                         int Nrows, int K, int OUT, int ldo) {
  const int wave = blockIdx.x * (blockDim.x >> 5) + (threadIdx.x >> 5);
  const int m0 = wave * 16;
  if (m0 >= Nrows) return;
  wave_linear_tile(A, W, bias, out, m0, Nrows, K, OUT, ldo, 0.01f, ACT);
}

// ---------------------------------------------------------------------------
// Per-node attention scalars: a_s[n] = hW[n]·as, a_d[n] = hW[n]·ad
// ---------------------------------------------------------------------------
__global__ void k_attn(const float* __restrict__ hW, const float* __restrict__ as_v,
                       const float* __restrict__ ad_v, float* __restrict__ a_s,
                       float* __restrict__ a_d, int Nrows) {
  const int wave = blockIdx.x * (blockDim.x >> 5) + (threadIdx.x >> 5);
  if (wave >= Nrows) return;
  const int lane = threadIdx.x & 31;
  const float* row = hW + (size_t)wave * 160;
  float ss = 0.0f, sd = 0.0f;
#pragma unroll
  for (int i = 0; i < 5; ++i) {
    const int k = lane + 32 * i;
    const float v = row[k];
    ss += v * as_v[k];
    sd += v * ad_v[k];
  }
#pragma unroll
  for (int off = 16; off; off >>= 1) {
    ss += __shfl_xor(ss, off, 32);
    sd += __shfl_xor(sd, off, 32);
  }
  if (lane == 0) { a_s[wave] = ss; a_d[wave] = sd; }
}

// ---------------------------------------------------------------------------
// Segment softmax over destinations + weighted scatter (3 edge passes)
// ---------------------------------------------------------------------------
__device__ __forceinline__ unsigned fkey(float f) {
  unsigned b = __float_as_uint(f);
  return (b & 0x80000000u) ? ~b : (b | 0x80000000u);
}
__device__ __forceinline__ float fkey_inv(unsigned k) {
  unsigned b = (k & 0x80000000u) ? (k & 0x7FFFFFFFu) : ~k;
  return __uint_as_float(b);
}

__global__ void k_init(float* __restrict__ out, const float* __restrict__ bias,
                       unsigned* __restrict__ m_u, float* __restrict__ z,
                       int Nrows, int D) {
  const int idx = blockIdx.x * blockDim.x + threadIdx.x;
  if (idx >= Nrows * D) return;
  const int n = idx / D, k = idx - n * D;
  out[idx] = bias[k];                   // pre-seed conv output with bias
  if (k == 0) { m_u[n] = 0u; z[n] = 0.0f; }
}

__global__ void k_edge_max(const int* __restrict__ ei, const float* __restrict__ a_s,
                           const float* __restrict__ a_d, float* __restrict__ e_buf,
                           unsigned* __restrict__ m_u, int E, int Nrows) {
  const int i = blockIdx.x * blockDim.x + threadIdx.x;
  if (i >= E + Nrows) return;
  int s, d;
  if (i < E) { s = ei[i]; d = ei[E + i]; } else { s = d = i - E; }
  float e = a_s[s] + a_d[d];
  e = e > 0.0f ? e : 0.2f * e;          // GAT leaky slope 0.2
  e_buf[i] = e;
  atomicMax(&m_u[d], fkey(e));
}

__global__ void k_edge_exp(const int* __restrict__ ei, float* __restrict__ e_buf,
                           const unsigned* __restrict__ m_u, float* __restrict__ z,
                           int E, int Nrows) {
  const int i = blockIdx.x * blockDim.x + threadIdx.x;
  if (i >= E + Nrows) return;
  int d;
  if (i < E) { d = ei[E + i]; } else { d = i - E; }
  const float ex = expf(e_buf[i] - fkey_inv(m_u[d]));
  e_buf[i] = ex;
  atomicAdd(&z[d], ex);
}

__global__ void k_edge_scatter(const int* __restrict__ ei, const float* __restrict__ e_buf,
                               const float* __restrict__ z, const float* __restrict__ hW,
                               float* __restrict__ out, int E, int Nrows) {
  const int idx = blockIdx.x * blockDim.x + threadIdx.x;
  if (idx >= (E + Nrows) * 40) return;
  const int i = idx / 40;               // edge id
  const int kc = (idx - i * 40) * 4;    // column chunk base
  int s, d;
  if (i < E) { s = ei[i]; d = ei[E + i]; } else { s = d = i - E; }
  const float alpha = e_buf[i] / (z[d] + 1e-16f);
  const float4 hv = *(const float4*)(hW + (size_t)s * 160 + kc);
  float* o = out + (size_t)d * 160 + kc;
  atomicAdd(o + 0, hv.x * alpha);
  atomicAdd(o + 1, hv.y * alpha);
  atomicAdd(o + 2, hv.z * alpha);
  atomicAdd(o + 3, hv.w * alpha);
}

// ---------------------------------------------------------------------------
// Final 80 -> 2 head (tiny, per-node)
// ---------------------------------------------------------------------------
__global__ void k_head2(const float* __restrict__ em, const float* __restrict__ w_o2,
                        const float* __restrict__ b_o2, float* __restrict__ out2,
                        int Nrows) {
  const int n = blockIdx.x * blockDim.x + threadIdx.x;
  if (n >= Nrows) return;
  float s0 = b_o2[0], s1 = b_o2[1];
  const float* r = em + (size_t)n * 80;
#pragma unroll 4
  for (int k = 0; k < 80; ++k) {
    const float v = r[k];
    s0 += v * w_o2[k];
    s1 += v * w_o2[80 + k];
  }
  out2[n * 2 + 0] = s0;
  out2[n * 2 + 1] = s1;
}

// ---------------------------------------------------------------------------
extern "C" void kernel_launch(void* const* d_in, const int* in_sizes, int n_in,
                              void* d_out, int out_size, void* d_ws, size_t ws_size,
                              hipStream_t stream) {
  const float* pre_x    = (const float*)d_in[0];
  const float* num_prop = (const float*)d_in[2];
  const float* num_cat  = (const float*)d_in[3];
  const float* des      = (const float*)d_in[4];
  const float* tweet    = (const float*)d_in[5];
  const int*   ei       = (const int*)d_in[6];
  const float* w_np  = (const float*)d_in[8];  const float* b_np  = (const float*)d_in[9];
  const float* w_nc  = (const float*)d_in[10]; const float* b_nc  = (const float*)d_in[11];
  const float* w_des = (const float*)d_in[12]; const float* b_des = (const float*)d_in[13];
  const float* w_text= (const float*)d_in[14]; const float* b_text= (const float*)d_in[15];
  const float* w_tw  = (const float*)d_in[16]; const float* b_tw  = (const float*)d_in[17];
  const float* w_in  = (const float*)d_in[18]; const float* b_in  = (const float*)d_in[19];
  const float* W1    = (const float*)d_in[20]; const float* b1    = (const float*)d_in[21];
  const float* as1   = (const float*)d_in[22]; const float* ad1   = (const float*)d_in[23];
  const float* W2    = (const float*)d_in[24]; const float* b2    = (const float*)d_in[25];
  const float* as2   = (const float*)d_in[26]; const float* ad2   = (const float*)d_in[27];
  const float* w_o1  = (const float*)d_in[28]; const float* b_o1  = (const float*)d_in[29];
  const float* w_o2  = (const float*)d_in[30]; const float* b_o2  = (const float*)d_in[31];

  const int N = in_sizes[0] / 768;
  const int E = in_sizes[7];
  const size_t nd = (size_t)N * 160;

  // Workspace layout (floats):
  float* ws   = (float*)d_ws;
  float* hcat = ws;             // h_cat, then reused as hW for both GAT layers
  float* hbuf = ws + nd;        // h after w_in, later layer-2 conv output
  float* g1   = ws + 2 * nd;    // layer-1 conv output
  float* a_s  = ws + 3 * nd;
  float* a_d  = a_s + N;
  unsigned* m_u = (unsigned*)(a_d + N);
  float* zbuf = (float*)(m_u + N);
  float* e_buf = zbuf + N;      // E + N entries

  float* logits = (float*)d_out;              // [N,2]
  float* em     = logits + (size_t)2 * N;     // [N,80]

  const int tiles = (N + 15) / 16;
  const dim3 blk(128);                         // 4 wave32 per block
  const int wblocks = (tiles + 3) / 4;
  const int Etot = E + N;
  const int nattn = (N + 3) / 4;
  const int ninit = (int)((nd + 255) / 256);
  const int nedge = (Etot + 255) / 256;
  const int nscat = (Etot * 40 + 255) / 256;

  // h_cat, then h = lrelu(h_cat @ w_in^T + b_in)
  k_fuse<<<wblocks, blk, 0, stream>>>(pre_x, num_prop, num_cat, des, tweet,
      w_np, b_np, w_nc, b_nc, w_des, b_des, w_text, b_text, w_tw, b_tw, hcat, N);
  k_linear<true><<<wblocks, blk, 0, stream>>>(hcat, w_in, b_in, hbuf, N, 160, 160, 160);

  // ---- GAT layer 1: hW = h @ W1^T ; softmax over dst ; g1 = scatter + b1
  k_linear<false><<<wblocks, blk, 0, stream>>>(hbuf, W1, nullptr, hcat, N, 160, 160, 160);
  k_attn<<<nattn, blk, 0, stream>>>(hcat, as1, ad1, a_s, a_d, N);
  k_init<<<ninit, 256, 0, stream>>>(g1, b1, m_u, zbuf, N, 160);
  k_edge_max<<<nedge, 256, 0, stream>>>(ei, a_s, a_d, e_buf, m_u, E, N);
  k_edge_exp<<<nedge, 256, 0, stream>>>(ei, e_buf, m_u, zbuf, E, N);
  k_edge_scatter<<<nscat, 256, 0, stream>>>(ei, e_buf, zbuf, hcat, g1, E, N);

  // ---- GAT layer 2 (no activation between convs): hbuf = conv2(g1)
  k_linear<false><<<wblocks, blk, 0, stream>>>(g1, W2, nullptr, hcat, N, 160, 160, 160);
  k_attn<<<nattn, blk, 0, stream>>>(hcat, as2, ad2, a_s, a_d, N);
  k_init<<<ninit, 256, 0, stream>>>(hbuf, b2, m_u, zbuf, N, 160);
  k_edge_max<<<nedge, 256, 0, stream>>>(ei, a_s, a_d, e_buf, m_u, E, N);
  k_edge_exp<<<nedge, 256, 0, stream>>>(ei, e_buf, m_u, zbuf, E, N);
  k_edge_scatter<<<nscat, 256, 0, stream>>>(ei, e_buf, zbuf, hcat, hbuf, E, N);

  // ---- head: em = lrelu(hbuf @ w_o1^T + b_o1) ; logits = em @ w_o2^T + b_o2
  k_linear<true><<<wblocks, blk, 0, stream>>>(hbuf, w_o1, b_o1, em, N, 160, 80, 80);
  k_head2<<<(N + 255) / 256, 256, 0, stream>>>(em, w_o2, b_o2, logits, N);
}